// PhysQuadModel_47553877902093
// MI455X (gfx1250) — compile-verified
//
#include <hip/hip_runtime.h>
#include <math.h>

// ---------------------------------------------------------------------------
// Quadrotor RK4 rollout, one thread per trajectory (wave32, gfx1250).
// State x = [pos(3), vel(3), so3(3), omega(3)] kept entirely in VGPRs.
// ---------------------------------------------------------------------------

namespace {

constexpr float kDT  = 0.01f;
constexpr float kM   = 0.033f;
constexpr float kG   = 9.81f;
constexpr float kT2W = 1.9f;
constexpr float kARM = 0.04f;
constexpr float kKT  = 3.8e-08f;
constexpr float kKC  = 3.8e-11f;
constexpr float kJx  = 1.4e-05f;
constexpr float kJy  = 1.4e-05f;
constexpr float kJz  = 2.17e-05f;
constexpr float kMTx = 0.009f;
constexpr float kMTy = 0.009f;
constexpr float kMTz = 0.002f;
constexpr float kTmax    = kT2W * kM * kG;
constexpr float kInvTmax = 1.0f / kTmax;
constexpr float kInvM    = 1.0f / kM;
constexpr float kInvJx   = 1.0f / kJx;
constexpr float kInvJy   = 1.0f / kJy;
constexpr float kInvJz   = 1.0f / kJz;
constexpr float kInvMTx  = 1.0f / kMTx;
constexpr float kInvMTy  = 1.0f / kMTy;
constexpr float kInvMTz  = 1.0f / kMTz;

struct V3 { float x, y, z; };
struct Q4 { float x, y, z, w; };  // (vector part, scalar part)

__device__ __forceinline__ V3 vadd(V3 a, V3 b) { return {a.x + b.x, a.y + b.y, a.z + b.z}; }
__device__ __forceinline__ V3 vscl(V3 a, float s) { return {a.x * s, a.y * s, a.z * s}; }
// b + s*a
__device__ __forceinline__ V3 vfma(V3 a, float s, V3 b) {
  return {fmaf(a.x, s, b.x), fmaf(a.y, s, b.y), fmaf(a.z, s, b.z)};
}
__device__ __forceinline__ float vdot(V3 a, V3 b) {
  return fmaf(a.x, b.x, fmaf(a.y, b.y, a.z * b.z));
}
__device__ __forceinline__ V3 vcross(V3 a, V3 b) {
  return {a.y * b.z - a.z * b.y, a.z * b.x - a.x * b.z, a.x * b.y - a.y * b.x};
}

// so3 log -> quaternion (faithful to reference, incl. small-angle branch)
__device__ __forceinline__ Q4 so3_to_quat(V3 r) {
  float th = sqrtf(vdot(r, r));
  float half = 0.5f * th;
  float s = __sinf(half);
  float c = __cosf(half);
  float k = s / (th + 1e-8f);
  bool small = th < 1e-6f;
  k = small ? 0.5f : k;
  float w = small ? 1.0f : c;
  return {k * r.x, k * r.y, k * r.z, w};
}

// quaternion -> so3 log (faithful: clamp w, eps=1e-6 small branch)
__device__ __forceinline__ V3 quat_to_so3(Q4 q) {
  float nv = sqrtf(q.x * q.x + q.y * q.y + q.z * q.z);
  float wc = fminf(fmaxf(q.w, -1.0f + 1e-6f), 1.0f - 1e-6f);
  float ang = 2.0f * atan2f(nv, wc);
  float k = ang / (nv + 1e-6f);
  k = (nv < 1e-6f) ? 2.0f : k;
  return {k * q.x, k * q.y, k * q.z};
}

__device__ __forceinline__ Q4 qnormalize(Q4 q) {
  float n = sqrtf(q.x * q.x + q.y * q.y + q.z * q.z + q.w * q.w);
  float inv = 1.0f / fmaxf(n, 1e-12f);
  return {q.x * inv, q.y * inv, q.z * inv, q.w * inv};
}

// dyn() with quat already materialized. T (newtons) and tau (N*m) pre-clamped.
// Returns acc, quat_dot, omega_dot. (pos-derivative == caller's vel.)
__device__ __forceinline__ void dyn_q(Q4 q, V3 omega, float T, V3 tau,
                                      V3& acc, Q4& qdot, V3& wdot) {
  V3 qv = {q.x, q.y, q.z};
  // thrust_w = quat_rotate(q, (0,0,T))
  V3 vb = {0.0f, 0.0f, T};
  V3 t2 = vscl(vcross(qv, vb), 2.0f);
  V3 tw = vadd(vadd(vb, vscl(t2, q.w)), vcross(qv, t2));
  acc = {tw.x * kInvM, tw.y * kInvM, fmaf(tw.z, kInvM, -kG)};
  // omega_dot = J^-1 (tau - omega x (J omega)), J diagonal
  V3 Jw = {kJx * omega.x, kJy * omega.y, kJz * omega.z};
  V3 c = vcross(omega, Jw);
  wdot = {(tau.x - c.x) * kInvJx, (tau.y - c.y) * kInvJy, (tau.z - c.z) * kInvJz};
  // quat_dot
  V3 dqv = vscl(vadd(vscl(omega, q.w), vcross(qv, omega)), 0.5f);
  qdot = {dqv.x, dqv.y, dqv.z, -0.5f * vdot(qv, omega)};
}

}  // namespace

__global__ __launch_bounds__(64, 1) void quad_rk4_rollout(
    const float4* __restrict__ x04,   // (B, 12) as 3 float4 per row
    const float4* __restrict__ u4,    // (B, N, 4) as 1 float4 per (b,t)
    float4* __restrict__ out4,        // (B, N, 12) as 3 float4 per (b,t)
    int B, int N) {
  int b = blockIdx.x * blockDim.x + threadIdx.x;
  if (b >= B) return;

  float4 xa = x04[b * 3 + 0];
  float4 xb = x04[b * 3 + 1];
  float4 xc = x04[b * 3 + 2];
  V3 pos   = {xa.x, xa.y, xa.z};
  V3 vel   = {xa.w, xb.x, xb.y};
  V3 so3   = {xb.z, xb.w, xc.x};
  V3 omega = {xc.y, xc.z, xc.w};

  const float4* __restrict__ ub = u4 + (size_t)b * (size_t)N;
  float4* __restrict__ ob = out4 + (size_t)b * (size_t)N * 3;

  const float h2 = 0.5f * kDT;
  const float k6 = kDT / 6.0f;

#pragma unroll 1
  for (int t = 0; t < N; ++t) {
    if (t + 8 < N) __builtin_prefetch(&ub[t + 8], 0, 1);  // global_prefetch_b8
    float4 u = ub[t];

    // motor_to_phys + clip, folded (clip(T_norm)=clip(T/Tmax), then *Tmax)
    float w0 = u.x * u.x, w1s = u.y * u.y, w2s = u.z * u.z, w3s = u.w * u.w;
    float Tn = kKT * (w0 + w1s + w2s + w3s) * kInvTmax;
    float txn = kKT * kARM * ((w2s + w3s) - (w0 + w1s)) * kInvMTx;
    float tyn = kKT * kARM * ((w1s + w2s) - (w0 + w3s)) * kInvMTy;
    float tzn = kKC * ((w0 + w2s) - (w1s + w3s)) * kInvMTz;
    float T = fminf(fmaxf(Tn, 0.0f), 1.0f) * kTmax;
    V3 tau = {fminf(fmaxf(txn, -1.0f), 1.0f) * kMTx,
              fminf(fmaxf(tyn, -1.0f), 1.0f) * kMTy,
              fminf(fmaxf(tzn, -1.0f), 1.0f) * kMTz};

    // ---- stage 1 (at x) ----
    Q4 q1 = so3_to_quat(so3);
    V3 a1, wd1; Q4 qd1;
    dyn_q(q1, omega, T, tau, a1, qd1, wd1);

    // ---- stage 2 (pack -> dyn, faithful quat->so3->quat roundtrip) ----
    Q4 qc2 = qnormalize({fmaf(h2, qd1.x, q1.x), fmaf(h2, qd1.y, q1.y),
                         fmaf(h2, qd1.z, q1.z), fmaf(h2, qd1.w, q1.w)});
    Q4 q2 = so3_to_quat(quat_to_so3(qc2));
    V3 om2 = vfma(wd1, h2, omega);
    V3 a2, wd2; Q4 qd2;
    dyn_q(q2, om2, T, tau, a2, qd2, wd2);

    // ---- stage 3 ----
    Q4 qc3 = qnormalize({fmaf(h2, qd2.x, q1.x), fmaf(h2, qd2.y, q1.y),
                         fmaf(h2, qd2.z, q1.z), fmaf(h2, qd2.w, q1.w)});
    Q4 q3 = so3_to_quat(quat_to_so3(qc3));
    V3 om3 = vfma(wd2, h2, omega);
    V3 a3, wd3; Q4 qd3;
    dyn_q(q3, om3, T, tau, a3, qd3, wd3);

    // ---- stage 4 ----
    Q4 qc4 = qnormalize({fmaf(kDT, qd3.x, q1.x), fmaf(kDT, qd3.y, q1.y),
                         fmaf(kDT, qd3.z, q1.z), fmaf(kDT, qd3.w, q1.w)});
    Q4 q4 = so3_to_quat(quat_to_so3(qc4));
    V3 om4 = vfma(wd3, kDT, omega);
    V3 a4, wd4; Q4 qd4;
    dyn_q(q4, om4, T, tau, a4, qd4, wd4);

    // ---- RK4 combine ----
    V3 v1 = vel;
    V3 v2 = vfma(a1, h2, vel);
    V3 v3 = vfma(a2, h2, vel);
    V3 v4 = vfma(a3, kDT, vel);
    pos   = vfma(vadd(vadd(v1, vscl(vadd(v2, v3), 2.0f)), v4), k6, pos);
    vel   = vfma(vadd(vadd(a1, vscl(vadd(a2, a3), 2.0f)), a4), k6, vel);
    omega = vfma(vadd(vadd(wd1, vscl(vadd(wd2, wd3), 2.0f)), wd4), k6, omega);
    Q4 qs = {qd1.x + 2.0f * (qd2.x + qd3.x) + qd4.x,
             qd1.y + 2.0f * (qd2.y + qd3.y) + qd4.y,
             qd1.z + 2.0f * (qd2.z + qd3.z) + qd4.z,
             qd1.w + 2.0f * (qd2.w + qd3.w) + qd4.w};
    Q4 qn = qnormalize({fmaf(k6, qs.x, q1.x), fmaf(k6, qs.y, q1.y),
                        fmaf(k6, qs.z, q1.z), fmaf(k6, qs.w, q1.w)});
    so3 = quat_to_so3(qn);

    // ---- emit x_next as 3x float4 (16B aligned: 48*t % 16 == 0) ----
    ob[t * 3 + 0] = make_float4(pos.x, pos.y, pos.z, vel.x);
    ob[t * 3 + 1] = make_float4(vel.y, vel.z, so3.x, so3.y);
    ob[t * 3 + 2] = make_float4(so3.z, omega.x, omega.y, omega.z);
  }
}

extern "C" void kernel_launch(void* const* d_in, const int* in_sizes, int n_in,
                              void* d_out, int out_size, void* d_ws, size_t ws_size,
                              hipStream_t stream) {
  (void)n_in; (void)out_size; (void)d_ws; (void)ws_size;
  const float4* x04 = (const float4*)d_in[0];   // (B,12) f32
  const float4* u4  = (const float4*)d_in[1];   // (B,N,4) f32
  float4* out4      = (float4*)d_out;           // (B,N,12) f32

  int B = in_sizes[0] / 12;
  int N = (B > 0) ? in_sizes[1] / (B * 4) : 0;

  const int threads = 64;                       // 2 waves/WG -> 128 WGs @ B=8192
  int blocks = (B + threads - 1) / threads;
  if (blocks > 0 && N > 0) {
    quad_rk4_rollout<<<blocks, threads, 0, stream>>>(x04, u4, out4, B, N);
  }
}